// nine_DoFAlignment_86792699118035
// MI455X (gfx1250) — compile-verified
//
#include <hip/hip_runtime.h>
#include <math.h>
#include <stdint.h>

// MI455X / gfx1250, wave32.
// Memory-bound: 201 MB @ 23.3 TB/s ~ 8.6 us floor.
// Kernel 1 (one block per batch):
//   - global_load_async_to_lds_b128 streams raw x[b], y[b] (24 KB) into LDS
//     (ASYNCcnt path, no VGPR round-trip, no staging VALU).
//   - V_WMMA_F32_16X16X4_F32 reduction; centering fused into the operand feed
//     as one FMA per element: a = fma(raw, mask, -mask*mu).
//     A rows 0-2 = xc, 3-5 = yc ; B cols 0-2 = yc, 3-5 = xc =>
//     D[0:3,0:3] = cov, diag D[0:3,3:6] = |xc|^2, diag D[3:6,0:3] = |yc|^2.
//   - writes cov(9) + xn2 + yn2 per batch to d_ws (stride 12 floats, 16B-aligned).
// Kernel 2 (one thread per batch): polar factor of cov (== U*Vt of its SVD) via
//   Frobenius-scaled Newton iteration; then t = mu_y - R*mu_x, scale = yn/(xn+1e-6).
// d_ws requirement: 8192 * 12 * 4 = 393216 bytes.

typedef __attribute__((ext_vector_type(2))) float v2f;
typedef __attribute__((ext_vector_type(4))) float v4f;
typedef __attribute__((ext_vector_type(8))) float v8f;

#define BATCH 8192
#define NPTS  1024

__device__ __forceinline__ float musel(int d, float m0, float m1, float m2) {
    return d == 0 ? m0 : (d == 1 ? m1 : m2);
}

__global__ __launch_bounds__(256) void ninedof_reduce(
    const float* __restrict__ x, const float* __restrict__ mu_x,
    const float* __restrict__ y, const float* __restrict__ mu_y,
    float* __restrict__ ws)
{
    __shared__ float sx[NPTS * 3];       // raw x tile, 12 KB
    __shared__ float sy[NPTS * 3];       // raw y tile, 12 KB
    __shared__ float spart[8 * 36];      // per-wave 6x6 partials
    __shared__ float sfinal[36];

    const int b   = blockIdx.x;
    const int tid = threadIdx.x;

    // Uniform (scalarized) mu loads
    const float mx0 = mu_x[b * 3 + 0], mx1 = mu_x[b * 3 + 1], mx2 = mu_x[b * 3 + 2];
    const float my0 = mu_y[b * 3 + 0], my1 = mu_y[b * 3 + 1], my2 = mu_y[b * 3 + 2];

    // ---- Async copy: 2 * 3072 floats = 2 * 768 b128 transfers, 3 each per thread.
    // LDS byte address for the async op = low 32 bits of the generic pointer.
    const float* gx = x + (size_t)b * (NPTS * 3);
    const float* gy = y + (size_t)b * (NPTS * 3);
    #pragma unroll
    for (int k = 0; k < 3; ++k) {
        const int i4 = (tid + k * 256) * 4;
        const unsigned lx = (unsigned)(uintptr_t)(&sx[i4]);
        const unsigned ly = (unsigned)(uintptr_t)(&sy[i4]);
        const unsigned long long ax = (unsigned long long)(uintptr_t)(gx + i4);
        const unsigned long long ay = (unsigned long long)(uintptr_t)(gy + i4);
        asm volatile("global_load_async_to_lds_b128 %0, %1, off"
                     :: "v"(lx), "v"(ax) : "memory");
        asm volatile("global_load_async_to_lds_b128 %0, %1, off"
                     :: "v"(ly), "v"(ay) : "memory");
    }
    asm volatile("s_wait_asynccnt 0x0" ::: "memory");
    __syncthreads();

    // ---- WMMA accumulation: 8 waves x 128 points, K=4 per WMMA -> 32 WMMAs/wave.
    const int wv   = tid >> 5;
    const int lane = tid & 31;
    const int half = lane >> 4;      // K-pair {0,1} vs {2,3}
    const int m    = lane & 15;      // M for A, N for B
    const float amask = (m < 6) ? 1.0f : 0.0f;
    const int   comp  = (m < 3) ? m : ((m < 6) ? (m - 3) : 0);
    const float* aBase = (m < 3) ? sx : sy;   // A rows 0-2 = xc, 3-5 = yc
    const float* bBase = (m < 3) ? sy : sx;   // B cols 0-2 = yc, 3-5 = xc

    const float mxc = musel(comp, mx0, mx1, mx2);
    const float myc = musel(comp, my0, my1, my2);
    const float nmu_a = -amask * ((m < 3) ? mxc : myc);
    const float nmu_b = -amask * ((m < 3) ? myc : mxc);

    v8f acc = {0.f, 0.f, 0.f, 0.f, 0.f, 0.f, 0.f, 0.f};
    const int nbase = wv * (NPTS / 8) + half * 2;
    #pragma unroll 4
    for (int it = 0; it < (NPTS / 8) / 4; ++it) {
        const int n0 = nbase + it * 4;
        const float a0 = fmaf(aBase[(n0    ) * 3 + comp], amask, nmu_a);
        const float a1 = fmaf(aBase[(n0 + 1) * 3 + comp], amask, nmu_a);
        const float b0 = fmaf(bBase[(n0    ) * 3 + comp], amask, nmu_b);
        const float b1 = fmaf(bBase[(n0 + 1) * 3 + comp], amask, nmu_b);
        v2f Am = {a0, a1};
        v2f Bm = {b0, b1};
        // D(16x16 f32) += A(16x4 f32) * B(4x16 f32)
        acc = __builtin_amdgcn_wmma_f32_16x16x4_f32(
            /*neg_a=*/false, Am, /*neg_b=*/false, Bm,
            /*c_mod=*/(short)0, acc, /*reuse_a=*/false, /*reuse_b=*/false);
    }

    // Extract 6x6 corner: lanes 0-5 hold N=lane, acc[r] is M=r (lanes < 16)
    if (lane < 6) {
        #pragma unroll
        for (int r = 0; r < 6; ++r)
            spart[wv * 36 + r * 6 + lane] = acc[r];
    }
    __syncthreads();

    if (tid < 36) {
        float s = 0.f;
        #pragma unroll
        for (int w = 0; w < 8; ++w) s += spart[w * 36 + tid];
        sfinal[tid] = s;
    }
    __syncthreads();

    float* wsb = ws + (size_t)b * 12;
    if (tid < 9) {
        const int Mi = tid / 3, Ni = tid % 3;
        wsb[tid] = sfinal[Mi * 6 + Ni];                       // cov[M][N]
    } else if (tid == 9) {
        wsb[9]  = sfinal[3]  + sfinal[10] + sfinal[17];       // xn^2
    } else if (tid == 10) {
        wsb[10] = sfinal[18] + sfinal[25] + sfinal[32];       // yn^2
    }
}

__global__ __launch_bounds__(256) void ninedof_finalize(
    const float* __restrict__ ws, const float* __restrict__ mu_x,
    const float* __restrict__ mu_y, float* __restrict__ out)
{
    const int b = blockIdx.x * blockDim.x + threadIdx.x;
    if (b >= BATCH) return;

    // 12-float record, 16B aligned -> three b128 loads
    const v4f* wv = (const v4f*)(ws + (size_t)b * 12);
    const v4f q0 = wv[0], q1 = wv[1], q2 = wv[2];
    float X[9] = {q0[0], q0[1], q0[2], q0[3], q1[0], q1[1], q1[2], q1[3], q2[0]};
    const float xn2 = q2[1], yn2 = q2[2];

    float fro2 = 0.f;
    #pragma unroll
    for (int i = 0; i < 9; ++i) fro2 += X[i] * X[i];
    const float fn = sqrtf(fro2);
    const float s0 = (fn > 1e-30f) ? (1.0f / fn) : 1.0f;
    #pragma unroll
    for (int i = 0; i < 9; ++i) X[i] *= s0;

    // Scaled Newton polar iteration: X <- 0.5*(zeta*X + (1/zeta)*X^-T)
    // -> orthogonal polar factor Q of cov == U*Vt from SVD.
    #pragma unroll
    for (int it = 0; it < 8; ++it) {
        float C[9];
        C[0] =   X[4] * X[8] - X[5] * X[7];
        C[1] = -(X[3] * X[8] - X[5] * X[6]);
        C[2] =   X[3] * X[7] - X[4] * X[6];
        C[3] = -(X[1] * X[8] - X[2] * X[7]);
        C[4] =   X[0] * X[8] - X[2] * X[6];
        C[5] = -(X[0] * X[7] - X[1] * X[6]);
        C[6] =   X[1] * X[5] - X[2] * X[4];
        C[7] = -(X[0] * X[5] - X[2] * X[3]);
        C[8] =   X[0] * X[4] - X[1] * X[3];
        const float det  = X[0] * C[0] + X[1] * C[1] + X[2] * C[2];
        const float ad   = fabsf(det);
        const float rdet = 1.0f / det;
        float fX2 = 0.f, fC2 = 0.f;
        #pragma unroll
        for (int i = 0; i < 9; ++i) { fX2 += X[i] * X[i]; fC2 += C[i] * C[i]; }
        const float zeta  = sqrtf(sqrtf(fC2) * fabsf(rdet) / sqrtf(fX2));
        const float izeta = rdet / zeta;   // (1/zeta)*(1/det) applied to cofactors (inv^T = cof/det)
        const bool  good  = (ad > 1e-20f);
        #pragma unroll
        for (int i = 0; i < 9; ++i) {
            const float xn = 0.5f * (zeta * X[i] + izeta * C[i]);
            X[i] = good ? xn : X[i];
        }
    }

    const float ux0 = mu_x[b * 3 + 0], ux1 = mu_x[b * 3 + 1], ux2 = mu_x[b * 3 + 2];
    const float uy0 = mu_y[b * 3 + 0], uy1 = mu_y[b * 3 + 1], uy2 = mu_y[b * 3 + 2];

    // Outputs: R [B,3,3] row-major, then t [B,3], then scale [B]
    #pragma unroll
    for (int i = 0; i < 9; ++i) out[(size_t)b * 9 + i] = X[i];

    out[(size_t)BATCH * 9 + b * 3 + 0] = uy0 - (X[0] * ux0 + X[1] * ux1 + X[2] * ux2);
    out[(size_t)BATCH * 9 + b * 3 + 1] = uy1 - (X[3] * ux0 + X[4] * ux1 + X[5] * ux2);
    out[(size_t)BATCH * 9 + b * 3 + 2] = uy2 - (X[6] * ux0 + X[7] * ux1 + X[8] * ux2);

    out[(size_t)BATCH * 12 + b] = sqrtf(yn2) / (sqrtf(xn2) + 1e-6f);
}

extern "C" void kernel_launch(void* const* d_in, const int* in_sizes, int n_in,
                              void* d_out, int out_size, void* d_ws, size_t ws_size,
                              hipStream_t stream) {
    const float* x    = (const float*)d_in[0];
    const float* mu_x = (const float*)d_in[1];
    const float* y    = (const float*)d_in[2];
    const float* mu_y = (const float*)d_in[3];
    float* ws  = (float*)d_ws;    // needs 8192*12*4 = 393216 bytes
    float* out = (float*)d_out;

    ninedof_reduce<<<BATCH, 256, 0, stream>>>(x, mu_x, y, mu_y, ws);
    ninedof_finalize<<<(BATCH + 255) / 256, 256, 0, stream>>>(ws, mu_x, mu_y, out);
}